// GAT_50431505990090
// MI455X (gfx1250) — compile-verified
//
#include <hip/hip_runtime.h>
#include <hip/hip_bf16.h>
#include <math.h>

#define NODES 50000
#define NEDGE 800000
#define ETOT  850000      // NEDGE + NODES self loops
#define DIM   128
#define HEADS 2
#define NB    128
#define NEG_SLOPE 0.2f

typedef float v2f __attribute__((ext_vector_type(2)));
typedef float v8f __attribute__((ext_vector_type(8)));

__device__ __forceinline__ void atomic_max_float(float* addr, float val) {
  // works because addr initialized to -inf; positives ordered as int, negatives reverse-ordered as uint
  if (val >= 0.0f) atomicMax((int*)addr, __float_as_int(val));
  else             atomicMin((unsigned int*)addr, __float_as_uint(val));
}
__device__ __forceinline__ float sigmoidf(float x) { return 1.0f / (1.0f + __expf(-x)); }

// ---------- pad raw features [N,9] -> [N,128] ----------
__global__ void k_pad(const float* __restrict__ x, float* __restrict__ h) {
  int t = blockIdx.x * blockDim.x + threadIdx.x;
  if (t >= NODES * DIM) return;
  int n = t >> 7, d = t & 127;
  h[t] = (d < 9) ? x[n * 9 + d] : 0.0f;
}

// ---------- XL[N,256] = H[N,128] @ W[128,256] via V_WMMA_F32_16X16X4_F32 ----------
// grid (3125, 4), block 128 (4 waves); wave w -> column tile blockIdx.y*4+w
__global__ void k_gemm_xl(const float* __restrict__ Hm, const float* __restrict__ W,
                          float* __restrict__ XL) {
  const int lane = threadIdx.x & 31;
  const int wave = threadIdx.x >> 5;
  const int row0 = blockIdx.x * 16;
  const int col0 = (blockIdx.y * 4 + wave) * 16;
  const int m  = lane & 15;       // A row / B col within tile
  const int kh = lane >> 4;       // K half select (lanes 16-31 -> K+2)
  v8f acc = {};
  const float* arow = Hm + (size_t)(row0 + m) * DIM;
  #pragma unroll 4
  for (int k0 = 0; k0 < DIM; k0 += 4) {
    const int ka = k0 + 2 * kh;
    v2f a; a.x = arow[ka];                     a.y = arow[ka + 1];
    v2f b; b.x = W[(size_t)ka * 256 + col0 + m]; b.y = W[(size_t)(ka + 1) * 256 + col0 + m];
    acc = __builtin_amdgcn_wmma_f32_16x16x4_f32(false, a, false, b, (short)0, acc,
                                                false, false);
  }
  // D layout: VGPR i, lane L -> (M = i + 8*(L>>4), N = L&15)
  float* orow = XL + (size_t)(row0 + 8 * kh) * 256 + col0 + m;
  #pragma unroll
  for (int i = 0; i < 8; ++i) orow[(size_t)i * 256] = acc[i];
}

// ---------- a_src/a_dst: one wave per (node, head) ----------
__global__ void k_attn(const float* __restrict__ XL, const float* __restrict__ att_src,
                       const float* __restrict__ att_dst,
                       float* __restrict__ Asrc, float* __restrict__ Adst) {
  int gid = blockIdx.x * blockDim.x + threadIdx.x;
  int pair = gid >> 5, lane = gid & 31;
  if (pair >= NODES * HEADS) return;
  int n = pair >> 1, h = pair & 1;
  const float* v  = XL + (size_t)n * 256 + h * 128;
  const float* as = att_src + h * 128;
  const float* ad = att_dst + h * 128;
  float s = 0.f, d = 0.f;
  #pragma unroll
  for (int i = 0; i < 4; ++i) {
    int c = lane + 32 * i;
    float xv = v[c];
    s += xv * as[c];
    d += xv * ad[c];
  }
  #pragma unroll
  for (int off = 16; off > 0; off >>= 1) {
    s += __shfl_down(s, off, 32);
    d += __shfl_down(d, off, 32);
  }
  if (lane == 0) { Asrc[pair] = s; Adst[pair] = d; }
}

// ---------- per-conv init: OUT=0, M=-inf, S=0 ----------
__global__ void k_conv_init(float* __restrict__ OUT, float* __restrict__ M, float* __restrict__ S) {
  int t = blockIdx.x * blockDim.x + threadIdx.x;
  if (t < NODES * 256) OUT[t] = 0.0f;
  if (t < NODES * HEADS) { M[t] = -INFINITY; S[t] = 0.0f; }
}

// ---------- edge pass 1: logits + segment max ----------
__global__ void k_edge_max(const long long* __restrict__ ei,
                           const float* __restrict__ Asrc, const float* __restrict__ Adst,
                           float* __restrict__ Elog, float* __restrict__ M) {
  int t = blockIdx.x * blockDim.x + threadIdx.x;
  if (t >= ETOT * HEADS) return;
  int e = t >> 1, h = t & 1;
  int src, dst;
  if (e < NEDGE) { src = (int)ei[e]; dst = (int)ei[NEDGE + e]; }
  else           { src = dst = e - NEDGE; }
  float l = Asrc[src * 2 + h] + Adst[dst * 2 + h];
  l = (l > 0.0f) ? l : NEG_SLOPE * l;               // leaky_relu
  Elog[t] = l;
  atomic_max_float(&M[dst * 2 + h], l);
}

// ---------- edge pass 2: exp + segment sum ----------
__global__ void k_edge_exp(const long long* __restrict__ ei, float* __restrict__ Elog,
                           const float* __restrict__ M, float* __restrict__ S) {
  int t = blockIdx.x * blockDim.x + threadIdx.x;
  if (t >= ETOT * HEADS) return;
  int e = t >> 1, h = t & 1;
  int dst = (e < NEDGE) ? (int)ei[NEDGE + e] : (e - NEDGE);
  float mm = M[dst * 2 + h];
  if (!(mm > -1e37f)) mm = 0.0f;                    // empty-segment guard (matches reference)
  float ex = __expf(Elog[t] - mm);
  Elog[t] = ex;
  atomicAdd(&S[dst * 2 + h], ex);
}

// ---------- edge pass 3: weighted scatter, one wave per (edge, head) ----------
__global__ void k_edge_scatter(const long long* __restrict__ ei, const float* __restrict__ XL,
                               const float* __restrict__ Elog, const float* __restrict__ S,
                               float* __restrict__ OUT) {
  int gid = blockIdx.x * blockDim.x + threadIdx.x;
  int w = gid >> 5, lane = gid & 31;
  if (w >= ETOT * HEADS) return;
  int e = w >> 1, h = w & 1;
  int src, dst;
  if (e < NEDGE) { src = (int)ei[e]; dst = (int)ei[NEDGE + e]; }
  else           { src = dst = e - NEDGE; }
  float alpha = Elog[w] / (S[dst * 2 + h] + 1e-16f);
  const float* xs = XL + (size_t)src * 256 + h * 128;
  float* od = OUT + (size_t)dst * 256 + h * 128;
  #pragma unroll
  for (int i = 0; i < 4; ++i) {
    int d = lane + 32 * i;
    atomicAdd(&od[d], alpha * xs[d]);               // GLOBAL_ATOMIC_ADD_F32, L2-resident
  }
}

// ---------- head mean + bias ----------
__global__ void k_head_mean(const float* __restrict__ OUT, const float* __restrict__ bias,
                            float* __restrict__ Hout) {
  int t = blockIdx.x * blockDim.x + threadIdx.x;
  if (t >= NODES * DIM) return;
  int n = t >> 7, d = t & 127;
  Hout[t] = 0.5f * (OUT[(size_t)n * 256 + d] + OUT[(size_t)n * 256 + 128 + d]) + bias[d];
}

// ---------- Set2Set ----------
__global__ void k_s2s_global_init(float* __restrict__ QSTAR, float* __restrict__ HST,
                                  float* __restrict__ CST) {
  int t = blockIdx.x * blockDim.x + threadIdx.x;
  if (t < NB * 256) QSTAR[t] = 0.0f;
  if (t < NB * 128) { HST[t] = 0.0f; CST[t] = 0.0f; }
}
__global__ void k_s2s_step_init(float* __restrict__ Bmax, float* __restrict__ Bsum,
                                float* __restrict__ R) {
  int t = blockIdx.x * blockDim.x + threadIdx.x;
  if (t < NB * DIM) R[t] = 0.0f;
  if (t < NB) { Bmax[t] = -INFINITY; Bsum[t] = 0.0f; }
}

// LSTM cell: one block per graph, 512 threads (one per gate element)
__global__ void k_lstm(const float* __restrict__ QSTAR, const float* __restrict__ Wih,
                       const float* __restrict__ Whh, const float* __restrict__ bih,
                       const float* __restrict__ bhh,
                       float* __restrict__ HST, float* __restrict__ CST) {
  __shared__ float sq[256];
  __shared__ float sh[128];
  __shared__ float sg[512];
  int b = blockIdx.x, j = threadIdx.x;
  if (j < 256) sq[j] = QSTAR[b * 256 + j];
  if (j < 128) sh[j] = HST[b * 128 + j];
  __syncthreads();
  float acc = bih[j] + bhh[j];
  const float* wi = Wih + (size_t)j * 256;
  for (int k = 0; k < 256; ++k) acc += sq[k] * wi[k];
  const float* wh = Whh + (size_t)j * 128;
  for (int k = 0; k < 128; ++k) acc += sh[k] * wh[k];
  sg[j] = acc;
  __syncthreads();
  if (j < 128) {
    float gi = sg[j], gf = sg[128 + j], gg = sg[256 + j], go = sg[384 + j];
    float c2 = sigmoidf(gf) * CST[b * 128 + j] + sigmoidf(gi) * tanhf(gg);
    float q  = sigmoidf(go) * tanhf(c2);
    CST[b * 128 + j] = c2;
    HST[b * 128 + j] = q;
  }
}

// e[n] = <x[n], q[batch[n]]>; one wave per node, also segment max
__global__ void k_s2s_energy(const float* __restrict__ Hn, const float* __restrict__ HST,
                             const long long* __restrict__ batch,
                             float* __restrict__ Enode, float* __restrict__ Bmax) {
  int gid = blockIdx.x * blockDim.x + threadIdx.x;
  int n = gid >> 5, lane = gid & 31;
  if (n >= NODES) return;
  int b = (int)batch[n];
  const float* xv = Hn + (size_t)n * 128;
  const float* qv = HST + (size_t)b * 128;
  float acc = 0.f;
  #pragma unroll
  for (int i = 0; i < 4; ++i) { int d = lane + 32 * i; acc += xv[d] * qv[d]; }
  #pragma unroll
  for (int off = 16; off > 0; off >>= 1) acc += __shfl_down(acc, off, 32);
  if (lane == 0) { Enode[n] = acc; atomic_max_float(&Bmax[b], acc); }
}

__global__ void k_s2s_exp(const long long* __restrict__ batch, float* __restrict__ Enode,
                          const float* __restrict__ Bmax, float* __restrict__ Bsum) {
  int n = blockIdx.x * blockDim.x + threadIdx.x;
  if (n >= NODES) return;
  int b = (int)batch[n];
  float mm = Bmax[b];
  if (!(mm > -1e37f)) mm = 0.0f;
  float ex = __expf(Enode[n] - mm);
  Enode[n] = ex;
  atomicAdd(&Bsum[b], ex);
}

__global__ void k_s2s_wsum(const long long* __restrict__ batch, const float* __restrict__ Hn,
                           const float* __restrict__ Enode, const float* __restrict__ Bsum,
                           float* __restrict__ R) {
  int gid = blockIdx.x * blockDim.x + threadIdx.x;
  int n = gid >> 5, lane = gid & 31;
  if (n >= NODES) return;
  int b = (int)batch[n];
  float a = Enode[n] / (Bsum[b] + 1e-16f);
  const float* xv = Hn + (size_t)n * 128;
  float* rb = R + (size_t)b * 128;
  #pragma unroll
  for (int i = 0; i < 4; ++i) { int d = lane + 32 * i; atomicAdd(&rb[d], a * xv[d]); }
}

__global__ void k_qstar(const float* __restrict__ HST, const float* __restrict__ R,
                        float* __restrict__ QSTAR) {
  int t = blockIdx.x * blockDim.x + threadIdx.x;
  if (t >= NB * 256) return;
  int b = t >> 8, j = t & 255;
  QSTAR[t] = (j < 128) ? HST[b * 128 + j] : R[b * 128 + (j - 128)];
}

// ---------- fused 2-layer MLP: one block per graph ----------
__global__ void k_mlp(const float* __restrict__ QSTAR, const float* __restrict__ W1,
                      const float* __restrict__ b1, const float* __restrict__ W2,
                      const float* __restrict__ b2, float* __restrict__ out) {
  __shared__ float sp[256];
  __shared__ float sh[128];
  int b = blockIdx.x, j = threadIdx.x;      // 128 threads
  sp[j]       = QSTAR[b * 256 + j];
  sp[128 + j] = QSTAR[b * 256 + 128 + j];
  __syncthreads();
  float acc = b1[j];
  for (int k = 0; k < 256; ++k) acc += sp[k] * W1[(size_t)k * 128 + j];
  sh[j] = fmaxf(acc, 0.0f);
  __syncthreads();
  float acc2 = b2[j];
  for (int k = 0; k < 128; ++k) acc2 += sh[k] * W2[(size_t)k * 128 + j];
  out[b * 128 + j] = acc2;
}

extern "C" void kernel_launch(void* const* d_in, const int* in_sizes, int n_in,
                              void* d_out, int out_size, void* d_ws, size_t ws_size,
                              hipStream_t stream) {
  (void)in_sizes; (void)n_in; (void)out_size; (void)ws_size;
  const float*     x        = (const float*)d_in[0];
  const long long* ei       = (const long long*)d_in[1];   // int64 [2,E]
  const long long* batch    = (const long long*)d_in[3];   // int64 [N]
  const float*     gat_W    = (const float*)d_in[4];
  const float*     att_src  = (const float*)d_in[5];
  const float*     att_dst  = (const float*)d_in[6];
  const float*     gat_bias = (const float*)d_in[7];
  const float*     Wih      = (const float*)d_in[8];
  const float*     Whh      = (const float*)d_in[9];
  const float*     bih      = (const float*)d_in[10];
  const float*     bhh      = (const float*)d_in[11];
  const float*     W1       = (const float*)d_in[12];
  const float*     b1       = (const float*)d_in[13];
  const float*     W2       = (const float*)d_in[14];
  const float*     b2       = (const float*)d_in[15];
  float* out = (float*)d_out;

  char* p = (char*)d_ws;
  auto carve = [&](size_t nelem) { float* r = (float*)p; p += nelem * sizeof(float); return r; };
  float* H0    = carve((size_t)NODES * DIM);
  float* H1    = carve((size_t)NODES * DIM);
  float* XL    = carve((size_t)NODES * 256);
  float* OUTB  = carve((size_t)NODES * 256);
  float* Asrc  = carve((size_t)NODES * HEADS);
  float* Adst  = carve((size_t)NODES * HEADS);
  float* M     = carve((size_t)NODES * HEADS);
  float* S     = carve((size_t)NODES * HEADS);
  float* Elog  = carve((size_t)ETOT * HEADS);
  float* Enode = carve((size_t)NODES);
  float* Bmax  = carve((size_t)NB);
  float* Bsum  = carve((size_t)NB);
  float* R     = carve((size_t)NB * DIM);
  float* QSTAR = carve((size_t)NB * 2 * DIM);
  float* HST   = carve((size_t)NB * DIM);
  float* CST   = carve((size_t)NB * DIM);

  const int TB = 256;
  auto cdiv = [](long long a, long long b) { return (unsigned)((a + b - 1) / b); };

  k_pad<<<cdiv((long long)NODES * DIM, TB), TB, 0, stream>>>(x, H0);

  float* Hin = H0; float* Hout = H1;
  for (int c = 0; c < 3; ++c) {
    k_gemm_xl<<<dim3(NODES / 16, 4), 128, 0, stream>>>(Hin, gat_W, XL);
    k_attn<<<cdiv((long long)NODES * HEADS * 32, TB), TB, 0, stream>>>(XL, att_src, att_dst, Asrc, Adst);
    k_conv_init<<<cdiv((long long)NODES * 256, TB), TB, 0, stream>>>(OUTB, M, S);
    k_edge_max<<<cdiv((long long)ETOT * HEADS, TB), TB, 0, stream>>>(ei, Asrc, Adst, Elog, M);
    k_edge_exp<<<cdiv((long long)ETOT * HEADS, TB), TB, 0, stream>>>(ei, Elog, M, S);
    k_edge_scatter<<<cdiv((long long)ETOT * HEADS * 32, TB), TB, 0, stream>>>(ei, XL, Elog, S, OUTB);
    k_head_mean<<<cdiv((long long)NODES * DIM, TB), TB, 0, stream>>>(OUTB, gat_bias, Hout);
    float* tmp = Hin; Hin = Hout; Hout = tmp;
  }

  k_s2s_global_init<<<cdiv(NB * 256, TB), TB, 0, stream>>>(QSTAR, HST, CST);
  for (int t = 0; t < 3; ++t) {
    k_lstm<<<NB, 512, 0, stream>>>(QSTAR, Wih, Whh, bih, bhh, HST, CST);
    k_s2s_step_init<<<cdiv(NB * DIM, TB), TB, 0, stream>>>(Bmax, Bsum, R);
    k_s2s_energy<<<cdiv((long long)NODES * 32, TB), TB, 0, stream>>>(Hin, HST, batch, Enode, Bmax);
    k_s2s_exp<<<cdiv(NODES, TB), TB, 0, stream>>>(batch, Enode, Bmax, Bsum);
    k_s2s_wsum<<<cdiv((long long)NODES * 32, TB), TB, 0, stream>>>(batch, Hin, Enode, Bsum, R);
    k_qstar<<<cdiv(NB * 256, TB), TB, 0, stream>>>(HST, R, QSTAR);
  }
  k_mlp<<<NB, 128, 0, stream>>>(QSTAR, W1, b1, W2, b2, out);
}